// KPMF_75668733821159
// MI455X (gfx1250) — compile-verified
//
#include <hip/hip_runtime.h>
#include <hip/hip_bf16.h>
#include <math.h>

// ---------------------------------------------------------------------------
// KPMF energy on MI455X (gfx1250, wave32, WMMA).
//
// E = SSE/(2 s^2) + 0.5*<S_U, U U^T> + 0.5*<S_V, V^T V> + N*log(s^2)
//
// Bandwidth bound: S_U + S_V = 512 MB streamed once (~22.5us @ 23.3 TB/s).
// Trace terms need 64 f32 MACs per streamed S element -> matrix pipe via
// V_WMMA_F32_16X16X4_F32 (full f32 precision). U (2MB) / V (2MB) stay
// L2-resident; S uses nontemporal loads so it does not evict them.
// S tile is preloaded into registers BEFORE the 16-WMMA chain so the global
// load latency overlaps the matrix ops. Reductions: LDS block reduce ->
// 256-way banked f32 atomics -> parallel weighted finalize.
// ---------------------------------------------------------------------------

typedef __attribute__((ext_vector_type(2))) float v2f;
typedef __attribute__((ext_vector_type(8))) float v8f;

#define KP_NROWS 8192
#define KP_NCOLS 8192
#define KP_RANK  64
#define KP_PAD   68   // LDS row pitch (floats): bank = 4*m (+hi) -> conflict-free
#define KP_BANKS 256  // accumulator banks per term

__global__ void kpmf_init_acc(float* __restrict__ acc) {
    int t = threadIdx.x;
    acc[t] = 0.0f;                 // SSE banks
    acc[KP_BANKS + t] = 0.0f;      // <S_U, UU^T> banks
    acc[2 * KP_BANKS + t] = 0.0f;  // <S_V, V^T V> banks
}

// ---------------------------------------------------------------------------
// Term 1: sum_n (vals - U[row]·V[:,col])^2  -> acc[0..255]
// ---------------------------------------------------------------------------
__global__ void kpmf_sparse_energy(const float* __restrict__ vals,
                                   const int*   __restrict__ rows,
                                   const int*   __restrict__ cols,
                                   const float* __restrict__ U,
                                   const float* __restrict__ V,
                                   float* __restrict__ acc, int n)
{
    __shared__ float red[8];
    int i = blockIdx.x * blockDim.x + threadIdx.x;
    float e = 0.0f;
    if (i < n) {
        int r = rows[i];
        int c = cols[i];
        const float4* u4 = reinterpret_cast<const float4*>(U + (size_t)r * KP_RANK);
        const float*  vc = V + c;
        float pred = 0.0f;
#pragma unroll
        for (int k = 0; k < KP_RANK / 4; ++k) {
            float4 u = u4[k];
            pred += u.x * vc[(size_t)(4 * k + 0) * KP_NCOLS];
            pred += u.y * vc[(size_t)(4 * k + 1) * KP_NCOLS];
            pred += u.z * vc[(size_t)(4 * k + 2) * KP_NCOLS];
            pred += u.w * vc[(size_t)(4 * k + 3) * KP_NCOLS];
        }
        float d = __builtin_nontemporal_load(&vals[i]) - pred;
        e = d * d;
    }
    // wave32 reduction
#pragma unroll
    for (int off = 16; off > 0; off >>= 1)
        e += __shfl_down(e, off, 32);
    if ((threadIdx.x & 31) == 0)
        red[threadIdx.x >> 5] = e;
    __syncthreads();
    if (threadIdx.x == 0) {
        float p = 0.0f;
#pragma unroll
        for (int w = 0; w < 8; ++w) p += red[w];
        atomicAdd(&acc[blockIdx.x & (KP_BANKS - 1)], p);
    }
}

// ---------------------------------------------------------------------------
// <S, F F^T> where F is 8192x64.
//   TRANS = false : F[n][r] = Fp[n*64 + r]        (U, row-major)
//   TRANS = true  : F[n][r] = Fp[r*8192 + n]      (V^T; V is 64x8192)
//
// Workgroup = 8 waves, owns a 128x64 super-tile of S (32 tiles of 16x16).
// Wave w: tile-row w, loops 4 J-tiles. Per 16x16 tile:
//   preload S tile (NT) -> G = F_I x F_J^T via 16x V_WMMA_F32_16X16X4_F32
//   (K=64) -> partial += sum(S_tile .* G).
// ---------------------------------------------------------------------------
template <bool TRANS>
__global__ void kpmf_gram_trace(const float* __restrict__ F,
                                const float* __restrict__ S,
                                float* __restrict__ acc)
{
    __shared__ float Fr[128][KP_PAD];   // rows i0..i0+127 of F
    __shared__ float Fc[64][KP_PAD];    // rows j0..j0+63  of F
    __shared__ float red[8];

    const int tid  = threadIdx.x;       // 0..255
    const int lane = tid & 31;
    const int wave = tid >> 5;          // 0..7
    const int i0   = blockIdx.x * 128;
    const int j0   = blockIdx.y * 64;

    // ---- stage factor blocks into LDS (coalesced global loads) ----
    if (!TRANS) {
        for (int t = tid; t < 128 * KP_RANK; t += 256) {
            int r = t & (KP_RANK - 1), n = t >> 6;
            Fr[n][r] = F[(size_t)(i0 + n) * KP_RANK + r];
        }
        for (int t = tid; t < 64 * KP_RANK; t += 256) {
            int r = t & (KP_RANK - 1), n = t >> 6;
            Fc[n][r] = F[(size_t)(j0 + n) * KP_RANK + r];
        }
    } else {
        for (int t = tid; t < 128 * KP_RANK; t += 256) {
            int n = t & 127, r = t >> 7;
            Fr[n][r] = F[(size_t)r * KP_NCOLS + (i0 + n)];
        }
        for (int t = tid; t < 64 * KP_RANK; t += 256) {
            int n = t & 63, r = t >> 6;
            Fc[n][r] = F[(size_t)r * KP_NCOLS + (j0 + n)];
        }
    }
    __syncthreads();

    const int m  = lane & 15;
    const int hi = lane >> 4;

    // A fragments for this wave's 16 rows, all 16 K-chunks (ISA 16x4 f32 layout:
    // lanes 0-15 hold K={0,1}, lanes 16-31 hold K={2,3})
    v2f a[16];
    const int iRow = wave * 16 + m;
#pragma unroll
    for (int k = 0; k < 16; ++k) {
        a[k].x = Fr[iRow][4 * k + 2 * hi + 0];
        a[k].y = Fr[iRow][4 * k + 2 * hi + 1];
    }

    float part = 0.0f;
    for (int jt = 0; jt < 4; ++jt) {
        // ---- issue all 8 NT loads of the S tile FIRST so their latency
        // overlaps the 16-WMMA chain (C/D layout: VGPR v, lane L ->
        // element [M = v + 8*(L>=16), N = L%16]) ----
        const size_t sBase = (size_t)(i0 + wave * 16) * KP_NCOLS + (j0 + jt * 16 + m);
        float sv[8];
#pragma unroll
        for (int v = 0; v < 8; ++v)
            sv[v] = __builtin_nontemporal_load(&S[sBase + (size_t)(v + 8 * hi) * KP_NCOLS]);

        // prefetch next S tile
        if (jt < 3) {
            const float* pf = S + (size_t)(i0 + wave * 16 + (lane >> 1)) * KP_NCOLS
                                + (j0 + (jt + 1) * 16 + (lane & 1) * 8);
            __builtin_prefetch(pf, 0, 1);
        }

        const int nCol = jt * 16 + m;
        v8f g = {};
#pragma unroll
        for (int k = 0; k < 16; ++k) {
            // B 4x16 fragment: rows striped across lanes within a VGPR
            v2f b;
            b.x = Fc[nCol][4 * k + hi];
            b.y = Fc[nCol][4 * k + 2 + hi];
            g = __builtin_amdgcn_wmma_f32_16x16x4_f32(
                    false, a[k], false, b, (short)0, g, false, false);
        }

        // dot the preloaded S tile against G
#pragma unroll
        for (int v = 0; v < 8; ++v)
            part += sv[v] * g[v];
    }

    // wave32 reduction -> block reduction -> one banked atomic per block
#pragma unroll
    for (int off = 16; off > 0; off >>= 1)
        part += __shfl_down(part, off, 32);
    if (lane == 0)
        red[wave] = part;
    __syncthreads();
    if (tid == 0) {
        float p = 0.0f;
#pragma unroll
        for (int w = 0; w < 8; ++w) p += red[w];
        int bank = (blockIdx.x * gridDim.y + blockIdx.y) & (KP_BANKS - 1);
        atomicAdd(&acc[bank], p);
    }
}

// ---------------------------------------------------------------------------
// out = sse/(2 s^2) + 0.5*(trU + trV) + N*log(s^2), 256-lane bank reduce.
// ---------------------------------------------------------------------------
__global__ void kpmf_finalize(const float* __restrict__ acc,
                              const float* __restrict__ sigma,
                              float* __restrict__ out, int n_obs)
{
    __shared__ float red[KP_BANKS];
    const int t = threadIdx.x;
    const float s2 = sigma[0] * sigma[0];
    float local = acc[t] / (2.0f * s2)
                + 0.5f * acc[KP_BANKS + t]
                + 0.5f * acc[2 * KP_BANKS + t];
    red[t] = local;
    __syncthreads();
#pragma unroll
    for (int off = KP_BANKS / 2; off > 0; off >>= 1) {
        if (t < off) red[t] += red[t + off];
        __syncthreads();
    }
    if (t == 0)
        out[0] = red[0] + (float)n_obs * logf(s2);
}

extern "C" void kernel_launch(void* const* d_in, const int* in_sizes, int n_in,
                              void* d_out, int out_size, void* d_ws, size_t ws_size,
                              hipStream_t stream)
{
    const float* vals  = (const float*)d_in[0];
    const int*   rows  = (const int*)d_in[1];
    const int*   cols  = (const int*)d_in[2];
    const float* U     = (const float*)d_in[3];
    const float* V     = (const float*)d_in[4];
    const float* sigma = (const float*)d_in[5];
    const float* S_U   = (const float*)d_in[6];
    const float* S_V   = (const float*)d_in[7];
    float* out = (float*)d_out;
    float* acc = (float*)d_ws;          // 3 * 256 banked partial sums
    const int n_obs = in_sizes[0];

    kpmf_init_acc<<<1, KP_BANKS, 0, stream>>>(acc);

    kpmf_sparse_energy<<<(n_obs + 255) / 256, 256, 0, stream>>>(
        vals, rows, cols, U, V, acc, n_obs);

    dim3 grid(KP_NROWS / 128, KP_NCOLS / 64);
    kpmf_gram_trace<false><<<grid, 256, 0, stream>>>(U, S_U, acc + KP_BANKS);
    kpmf_gram_trace<true ><<<grid, 256, 0, stream>>>(V, S_V, acc + 2 * KP_BANKS);

    kpmf_finalize<<<1, KP_BANKS, 0, stream>>>(acc, sigma, out, n_obs);
}